// Net_11587821765063
// MI455X (gfx1250) — compile-verified
//
#include <hip/hip_runtime.h>
#include <hip/hip_bf16.h>

// ---------------- problem constants ----------------
#define T_STEPS 1000
#define CIN     784
#define COUT    512
#define KPAD    896            // 56 k-tiles of 16 (CIN padded; pad cols stay 0)
#define NWG     32             // each WG owns 16 output rows (one WMMA m-tile)
#define NTHR    256            // 8 wave32s per WG
#define NWAVE   8
#define TPW     7              // 56 k-tiles / 8 waves
#define VTHRv   12500.0f
#define PROHIBv 11250.0f

typedef float v2f __attribute__((ext_vector_type(2)));
typedef float v8f __attribute__((ext_vector_type(8)));
typedef int   v4i __attribute__((vector_size(16)));   // matches builtin's V4i

// ---------------- CDNA5 async global->LDS (guarded) ----------------
#if defined(__HIP_DEVICE_COMPILE__) && __has_builtin(__builtin_amdgcn_global_load_async_to_lds_b128)
#define HAVE_ASYNC_LDS 1
#else
#define HAVE_ASYNC_LDS 0
#endif

__device__ __forceinline__ void async_copy_b128(const void* gsrc, void* ldst) {
#if HAVE_ASYNC_LDS
  __builtin_amdgcn_global_load_async_to_lds_b128(
      (__attribute__((address_space(1))) v4i*)gsrc,
      (__attribute__((address_space(3))) v4i*)ldst,
      /*offset=*/0, /*cpol=*/0);
#else
  *(int4*)ldst = *(const int4*)gsrc;   // synchronous fallback
#endif
}

__device__ __forceinline__ void wait_async_all() {
#if HAVE_ASYNC_LDS
#if __has_builtin(__builtin_amdgcn_s_wait_asynccnt)
  __builtin_amdgcn_s_wait_asynccnt(0);
#else
  asm volatile("s_wait_asynccnt 0" ::: "memory");
#endif
#endif
}

// LUT[27..31] = {0,-1,2,1,0}; everything else 0  (reference _raw_lut)
__device__ __forceinline__ float lutval(int idx) {
  if (idx == 28) return -1.f;
  if (idx == 29) return  2.f;
  if (idx == 30) return  1.f;
  return 0.f;
}

__global__ void init_flags(int* f, int n) {
  int i = blockIdx.x * blockDim.x + threadIdx.x;
  if (i < n) f[i] = 0;
}

// Persistent SNN kernel: W^T held in WMMA C/D register tiles for all 1000 steps.
__global__ void __launch_bounds__(NTHR)
snn_kernel(const int* __restrict__ x, const float* __restrict__ weight,
           float* __restrict__ out, int* __restrict__ gAny, int* __restrict__ gDone) {
  __shared__ __align__(16) float s_xs[KPAD];     // xt as f32 (pad = 0)
  __shared__ float s_gin[KPAD];                  // gate_in * LUT[cin]  (pad = 0)
  __shared__ int   s_cin[CIN];
  __shared__ __align__(16) int s_xraw[2][CIN];   // double-buffered async stage
  __shared__ float s_part[NWAVE * 16];
  __shared__ float s_mem[16];
  __shared__ float s_sp[16];
  __shared__ float s_gout[16];
  __shared__ int   s_cout[16];
  __shared__ int   s_anyPrev;

  const int tid  = threadIdx.x;
  const int lane = tid & 31;
  const int wv   = tid >> 5;        // wave id 0..7
  const int half = lane >> 4;       // C/D layout: lanes 16..31 hold M(+8)
  const int mloc = lane & 15;       // output-neuron column of this lane
  const int wg   = blockIdx.x;
  const int mg   = wg * 16 + mloc;  // global output row owned by this lane

  // ---- init per-WG state ----
  for (int i = tid; i < KPAD; i += NTHR) { s_xs[i] = 0.f; s_gin[i] = 0.f; }
  for (int i = tid; i < CIN;  i += NTHR) s_cin[i] = 0;
  if (tid < 16) { s_mem[tid] = 0.f; s_cout[tid] = 0; s_sp[tid] = 0.f; s_gout[tid] = 0.f; }

  // ---- load this lane's slice of W^T into 7 WMMA accumulator tiles ----
  // tile j covers k in [ (wv*7+j)*16 , +16 ); lane holds k = base + half*8 + v
  v8f wreg[TPW];
#pragma unroll
  for (int j = 0; j < TPW; ++j) {
    const int kb = (wv * TPW + j) * 16 + half * 8;
    if (kb + 7 < CIN) {
      const float4 a = *(const float4*)&weight[(size_t)mg * CIN + kb];
      const float4 b = *(const float4*)&weight[(size_t)mg * CIN + kb + 4];
      wreg[j][0]=a.x; wreg[j][1]=a.y; wreg[j][2]=a.z; wreg[j][3]=a.w;
      wreg[j][4]=b.x; wreg[j][5]=b.y; wreg[j][6]=b.z; wreg[j][7]=b.w;
    } else {
      wreg[j] = 0.f;                 // padded k-columns stay zero forever
    }
  }

  // ---- stage x row 0 into LDS (async if available) ----
  if (tid < CIN / 4) async_copy_b128(&x[tid * 4], &s_xraw[0][tid * 4]);

  for (int t = 0; t < T_STEPS; ++t) {
    wait_async_all();                              // x[t] resident in LDS
    const int tn = t + 1;                          // prefetch x[t+1]
    if (tn < T_STEPS && tid < CIN / 4)
      async_copy_b128(&x[(size_t)tn * CIN + tid * 4], &s_xraw[tn & 1][tid * 4]);
    __syncthreads();                               // also orders prev stage D vs new xs

    // ---- stage A: input-side state (replicated per WG) + inter-WG sync ----
    if (tid == 0) {
      int ap = 0;
      if (t > 0) {
        while (atomicAdd(&gDone[t - 1], 0) < NWG) __builtin_amdgcn_s_sleep(1);
        __threadfence();
        ap = atomicAdd(&gAny[t - 1], 0);           // any spike at t-1, whole net
      }
      s_anyPrev = ap;
    }
    for (int k = tid; k < CIN; k += NTHR) {
      const int xv  = s_xraw[t & 1][k];
      const int cn  = (xv == 1) ? 0 : min(s_cin[k] + 1, 31);
      s_cin[k] = cn;
      s_xs[k]  = (float)xv;
      const int idx = min(cn + 29, 59);            // SHIFT=29, clip high
      s_gin[k] = (cn <= 30) ? lutval(idx) : 0.f;   // gate_in * lut_in
    }
    __syncthreads();

    // ---- stage B: mem += W @ xt  (per-lane FMAs on register tiles) ----
    float acc = 0.f;
#pragma unroll
    for (int j = 0; j < TPW; ++j) {
      const int kb = (wv * TPW + j) * 16 + half * 8;
      const float4 xa = *(const float4*)&s_xs[kb];
      const float4 xb = *(const float4*)&s_xs[kb + 4];
      acc += wreg[j][0]*xa.x + wreg[j][1]*xa.y + wreg[j][2]*xa.z + wreg[j][3]*xa.w;
      acc += wreg[j][4]*xb.x + wreg[j][5]*xb.y + wreg[j][6]*xb.z + wreg[j][7]*xb.w;
    }
    acc += __shfl_xor(acc, 16, 32);                // merge lane-halves (k +8)
    if (lane < 16) s_part[wv * 16 + lane] = acc;
    __syncthreads();

    // ---- stage C: threshold, reset, output, cout/gout, global any-spike ----
    if (tid < 16) {
      float y = s_mem[tid];
#pragma unroll
      for (int w8 = 0; w8 < NWAVE; ++w8) y += s_part[w8 * 16 + tid];
      if (s_anyPrev) y -= PROHIBv;                 // prohibit from t-1 spikes
      y = fmaxf(y, 0.f);
      const bool  spike = (y >= VTHRv);
      s_mem[tid] = spike ? 0.f : y;
      const float spf = spike ? 1.f : 0.f;
      s_sp[tid] = spf;
      out[(size_t)t * COUT + wg * 16 + tid] = spf;
      const int cn  = spike ? 0 : min(s_cout[tid] + 1, 31);
      s_cout[tid] = cn;
      const int idx = max(29 - cn, 0);             // clip low
      s_gout[tid] = (1.f - spf) * ((cn <= 29) ? lutval(idx) : 0.f);
      const int any = __any(spike);
      if (tid == 0) {
        if (any) atomicOr(&gAny[t], 1);
        __threadfence();
        atomicAdd(&gDone[t], 1);                   // release arrival for step t
      }
    }
    __syncthreads();

    // ---- stage D: plasticity as rank-1 WMMAs on the resident W^T tiles ----
    // change1: dW^T[k,m] = gin[k] * sp[m];  clamp min(.,127)
    // change2: dW^T[k,m] = x[k] * gout[m];  clamp max(.,0)
    // A(16x4): lanes 0-15 vgpr0 = column K=0; everything else zero.
    // B(4x16): vgpr0 lanes 0-15 = row K=0; everything else zero.
#pragma unroll
    for (int j = 0; j < TPW; ++j) {
      const int kb = (wv * TPW + j) * 16;
      v2f A, B;
      A[0] = (lane < 16) ? s_gin[kb + lane] : 0.f;  A[1] = 0.f;
      B[0] = (lane < 16) ? s_sp[lane]       : 0.f;  B[1] = 0.f;
      wreg[j] = __builtin_amdgcn_wmma_f32_16x16x4_f32(
          false, A, false, B, (short)0, wreg[j], false, false);
#pragma unroll
      for (int v = 0; v < 8; ++v) wreg[j][v] = fminf(wreg[j][v], 127.f);
      A[0] = (lane < 16) ? s_xs[kb + lane]  : 0.f;
      B[0] = (lane < 16) ? s_gout[lane]     : 0.f;
      wreg[j] = __builtin_amdgcn_wmma_f32_16x16x4_f32(
          false, A, false, B, (short)0, wreg[j], false, false);
#pragma unroll
      for (int v = 0; v < 8; ++v) wreg[j][v] = fmaxf(wreg[j][v], 0.f);
    }
    // top-of-loop __syncthreads orders stage D against next step's stage A
  }
}

extern "C" void kernel_launch(void* const* d_in, const int* in_sizes, int n_in,
                              void* d_out, int out_size, void* d_ws, size_t ws_size,
                              hipStream_t stream) {
  const int*   x = (const int*)d_in[0];      // (T,1,CIN) int32
  const float* w = (const float*)d_in[1];    // (COUT,CIN) f32
  float* out = (float*)d_out;                // (T,1,COUT) f32 spikes
  int* flags = (int*)d_ws;                   // [0,T): gAny, [T,2T): gDone
  init_flags<<<(2 * T_STEPS + 255) / 256, 256, 0, stream>>>(flags, 2 * T_STEPS);
  snn_kernel<<<NWG, NTHR, 0, stream>>>(x, w, out, flags, flags + T_STEPS);
}